// TDE_70935679861198
// MI455X (gfx1250) — compile-verified
//
#include <hip/hip_runtime.h>

typedef float v2f __attribute__((ext_vector_type(2)));
typedef float v8f __attribute__((ext_vector_type(8)));

#define BS      512
#define NF      128        // nf = ntips
#define NROW    254        // N = 2n-2
#define DIMV    126        // valid internal rows (padded to 128)
#define NPAD    256        // padded columns of the bottom block
#define T_MAX   20         // A^(2^21-2): far beyond convergence of the doubling
#define TOL_SUM 0.32512f   // 1e-5 * 254 * 128  (sum-of-|diff| threshold == mean > TOL)

#define AB_ELEMS (128L * 256L)   // bottom-block elems per batch
#define XB_ELEMS (128L * 128L)   // Xb elems per batch
#define G2_BLOCKS 2              // gemm2 blocks per batch (partial-sum stride)

#define WMMA_F32(a, bfrag, c) \
    __builtin_amdgcn_wmma_f32_16x16x4_f32(false, (a), false, (bfrag), (short)0, (c), false, false)

// ---------------- build kernels ----------------

// Dense-build padded bottom block: Ab[b][i][j] (i<128, j<256)
__global__ void k_build_A(const int* __restrict__ ei, float* __restrict__ Ab) {
    int b = blockIdx.y;
    int t = blockIdx.x * blockDim.x + threadIdx.x;   // 0 .. 32767
    int i = t >> 8;                                  // bottom row 0..127
    int j = t & 255;                                 // col 0..255
    float v = 0.0f;
    if (i < DIMV) {
        const int* e = ei + ((long)b * NROW + (NF + i)) * 3;
        int e0 = e[0], e1 = e[1], e2 = e[2];
        if (j == e0 || j == e1 || j == e2) v = 1.0f / 3.0f;
    }
    Ab[((long)b << 15) + t] = v;
}

// Xb0[b][i][j] = x_init[b][128+i][j] for i<126, pad rows 0
__global__ void k_build_X(const float* __restrict__ xinit, float* __restrict__ Xb) {
    int b = blockIdx.y;
    int t = blockIdx.x * blockDim.x + threadIdx.x;   // 0 .. 16383
    int i = t >> 7;
    int j = t & 127;
    float v = 0.0f;
    if (i < DIMV) v = xinit[((long)b * NROW + (NF + i)) * NF + j];
    Xb[((long)b << 14) + t] = v;
}

__global__ void k_init_flags(int* conv, int* fidx) {
    if (threadIdx.x == 0) { *conv = 0; *fidx = 0; }
}

// ---------------- GEMM 1: Anew = [L,0] + R @ Acur (128x256, K=128), 32x32 per wave ----------------
__global__ void __launch_bounds__(256) k_gemm1(const float* __restrict__ Acur,
                                               float* __restrict__ Anew,
                                               const int* __restrict__ conv) {
    if (*(volatile const int*)conv) return;
    int b = blockIdx.y;
    const float* A = Acur + ((long)b << 15);
    float*       D = Anew + ((long)b << 15);

    int wave = threadIdx.x >> 5;
    int lane = threadIdx.x & 31;
    int sup  = blockIdx.x * 8 + wave;      // 0..31 : 4 super-rows x 8 super-cols of 32x32
    int row0 = (sup >> 3) * 32;
    int col0 = (sup & 7) * 32;
    int ml = lane & 15, hi = lane >> 4, kk = hi * 2;
    int c0 = col0 + ml, c1 = c0 + 16;

    v8f acc00, acc01, acc10, acc11;
#pragma unroll
    for (int v = 0; v < 8; ++v) {          // bias = [L, 0]
        int r0 = row0 + v + 8 * hi, r1 = r0 + 16;
        acc00[v] = (c0 < NF) ? A[r0 * NPAD + c0] : 0.0f;
        acc01[v] = (c1 < NF) ? A[r0 * NPAD + c1] : 0.0f;
        acc10[v] = (c0 < NF) ? A[r1 * NPAD + c0] : 0.0f;
        acc11[v] = (c1 < NF) ? A[r1 * NPAD + c1] : 0.0f;
    }

    const float* Ar0 = A + (row0 + ml) * NPAD + NF;   // R-block rows of this lane
    const float* Ar1 = Ar0 + 16 * NPAD;
#pragma unroll 4
    for (int k = 0; k < NF; k += 4) {
        v2f a0 = *reinterpret_cast<const v2f*>(Ar0 + k + kk);
        v2f a1 = *reinterpret_cast<const v2f*>(Ar1 + k + kk);
        const float* Br0 = A + (k + kk) * NPAD;
        const float* Br1 = Br0 + NPAD;
        v2f b0; b0.x = Br0[c0]; b0.y = Br1[c0];
        v2f b1; b1.x = Br0[c1]; b1.y = Br1[c1];
        acc00 = WMMA_F32(a0, b0, acc00);
        acc01 = WMMA_F32(a0, b1, acc01);
        acc10 = WMMA_F32(a1, b0, acc10);
        acc11 = WMMA_F32(a1, b1, acc11);
    }

#pragma unroll
    for (int v = 0; v < 8; ++v) {
        int r0 = row0 + v + 8 * hi, r1 = r0 + 16;
        D[r0 * NPAD + c0] = acc00[v];
        D[r0 * NPAD + c1] = acc01[v];
        D[r1 * NPAD + c0] = acc10[v];
        D[r1 * NPAD + c1] = acc11[v];
    }
}

// ---------------- GEMM 2: Xn = L' + R' @ Xo (128x128, K=128), 32x32 per wave + |diff| sums -------
__global__ void __launch_bounds__(256) k_gemm2(const float* __restrict__ Bn,
                                               const float* __restrict__ Xo,
                                               float* __restrict__ Xn,
                                               float* __restrict__ parts,
                                               const int* __restrict__ conv) {
    if (*(volatile const int*)conv) return;
    int b = blockIdx.y;
    const float* B = Bn + ((long)b << 15);
    const float* X = Xo + ((long)b << 14);
    float*       D = Xn + ((long)b << 14);

    int wave = threadIdx.x >> 5;
    int lane = threadIdx.x & 31;
    int sup  = blockIdx.x * 8 + wave;      // 0..15 : 4 super-rows x 4 super-cols of 32x32
    int row0 = (sup >> 2) * 32;
    int col0 = (sup & 3) * 32;
    int ml = lane & 15, hi = lane >> 4, kk = hi * 2;
    int c0 = col0 + ml, c1 = c0 + 16;

    v8f acc00, acc01, acc10, acc11;
#pragma unroll
    for (int v = 0; v < 8; ++v) {          // bias = L'
        int r0 = row0 + v + 8 * hi, r1 = r0 + 16;
        acc00[v] = B[r0 * NPAD + c0];
        acc01[v] = B[r0 * NPAD + c1];
        acc10[v] = B[r1 * NPAD + c0];
        acc11[v] = B[r1 * NPAD + c1];
    }

    const float* Br0 = B + (row0 + ml) * NPAD + NF;   // R'-block rows
    const float* Br1 = Br0 + 16 * NPAD;
#pragma unroll 4
    for (int k = 0; k < NF; k += 4) {
        v2f a0 = *reinterpret_cast<const v2f*>(Br0 + k + kk);
        v2f a1 = *reinterpret_cast<const v2f*>(Br1 + k + kk);
        const float* Xr0 = X + (k + kk) * NF;
        const float* Xr1 = Xr0 + NF;
        v2f b0; b0.x = Xr0[c0]; b0.y = Xr1[c0];
        v2f b1; b1.x = Xr0[c1]; b1.y = Xr1[c1];
        acc00 = WMMA_F32(a0, b0, acc00);
        acc01 = WMMA_F32(a0, b1, acc01);
        acc10 = WMMA_F32(a1, b0, acc10);
        acc11 = WMMA_F32(a1, b1, acc11);
    }

    float s = 0.0f;
#pragma unroll
    for (int v = 0; v < 8; ++v) {
        int r0 = row0 + v + 8 * hi, r1 = r0 + 16;
        float o00 = X[r0 * NF + c0], o01 = X[r0 * NF + c1];
        float o10 = X[r1 * NF + c0], o11 = X[r1 * NF + c1];
        D[r0 * NF + c0] = acc00[v];
        D[r0 * NF + c1] = acc01[v];
        D[r1 * NF + c0] = acc10[v];
        D[r1 * NF + c1] = acc11[v];
        s += fabsf(acc00[v] - o00) + fabsf(acc01[v] - o01) +
             fabsf(acc10[v] - o10) + fabsf(acc11[v] - o11);
    }

    // deterministic block reduction (LDS tree), no float atomics
    __shared__ float red[256];
    red[threadIdx.x] = s;
    __syncthreads();
#pragma unroll
    for (int off = 128; off > 0; off >>= 1) {
        if (threadIdx.x < off) red[threadIdx.x] += red[threadIdx.x + off];
        __syncthreads();
    }
    if (threadIdx.x == 0) parts[b * G2_BLOCKS + blockIdx.x] = red[0];
}

// ---------------- convergence check ----------------
__global__ void k_check(const float* __restrict__ parts, int* conv, int* fidx, int pout) {
    if (*(volatile const int*)conv) return;
    __shared__ int s_any;
    if (threadIdx.x == 0) s_any = 0;
    __syncthreads();
    float s = 0.0f;
#pragma unroll
    for (int i = 0; i < G2_BLOCKS; ++i) s += parts[threadIdx.x * G2_BLOCKS + i];
    if (s > TOL_SUM) s_any = 1;           // benign race, all write 1
    __syncthreads();
    if (threadIdx.x == 0) {
        *fidx = pout;                      // buffer holding last-written Xb
        if (!s_any) *conv = 1;
    }
}

// ---------------- finalize: out = [x_init_top ; Xb_final] ----------------
__global__ void k_final(const float* __restrict__ xinit, const float* __restrict__ X0,
                        const float* __restrict__ X1, const int* __restrict__ fidx,
                        float* __restrict__ out) {
    long idx = (long)blockIdx.x * blockDim.x + threadIdx.x;
    const long TOT = (long)BS * NROW * NF;
    if (idx >= TOT) return;
    long col = idx & (NF - 1);
    long rem = idx >> 7;                  // b*254 + row
    int  row = (int)(rem % NROW);
    long b   = rem / NROW;
    float v;
    if (row < NF) {
        v = xinit[idx];                   // identity top passes through exactly
    } else {
        const float* X = (*fidx) ? X1 : X0;
        v = X[b * XB_ELEMS + (long)(row - NF) * NF + col];
    }
    out[idx] = v;
}

extern "C" void kernel_launch(void* const* d_in, const int* in_sizes, int n_in,
                              void* d_out, int out_size, void* d_ws, size_t ws_size,
                              hipStream_t stream) {
    const int*   ei    = (const int*)d_in[0];
    const float* xinit = (const float*)d_in[1];
    float* out = (float*)d_out;

    float* ws = (float*)d_ws;
    float* Ab0 = ws;
    float* Ab1 = Ab0 + (long)BS * AB_ELEMS;
    float* Xb0 = Ab1 + (long)BS * AB_ELEMS;
    float* Xb1 = Xb0 + (long)BS * XB_ELEMS;
    float* parts = Xb1 + (long)BS * XB_ELEMS;       // BS*G2_BLOCKS partial sums
    int*   conv  = (int*)(parts + (long)BS * G2_BLOCKS);
    int*   fidx  = conv + 1;
    float* AbP[2] = {Ab0, Ab1};
    float* XbP[2] = {Xb0, Xb1};

    k_build_A<<<dim3(128, BS), 256, 0, stream>>>(ei, Ab0);
    k_build_X<<<dim3(64, BS), 256, 0, stream>>>(xinit, Xb0);
    k_init_flags<<<1, 32, 0, stream>>>(conv, fidx);

    for (int t = 0; t < T_MAX; ++t) {
        int cur = t & 1, nxt = cur ^ 1;
        k_gemm1<<<dim3(4, BS), 256, 0, stream>>>(AbP[cur], AbP[nxt], conv);
        k_gemm2<<<dim3(G2_BLOCKS, BS), 256, 0, stream>>>(AbP[nxt], XbP[cur], XbP[nxt], parts, conv);
        k_check<<<1, BS, 0, stream>>>(parts, conv, fidx, nxt);
    }

    long total = (long)BS * NROW * NF;
    int blocks = (int)((total + 255) / 256);
    k_final<<<blocks, 256, 0, stream>>>(xinit, Xb0, Xb1, fidx, out);
}